// TensorBoard_4423816315108
// MI455X (gfx1250) — compile-verified
//
#include <hip/hip_runtime.h>
#include <hip/hip_bf16.h>

// ---------------------------------------------------------------------------
// Go superko repeat-check (B=1024 boards, 19x19, M=722 history entries).
//
// MI455X reasoning: 267M int32 equality tests over ~10MB of data -> latency
// bound (0.45us of HBM traffic at 23.3 TB/s). Equality matching is not a
// matmul, so WMMA is the wrong tool (an IU8 distance-trick WMMA resolves 256
// pairs per 32K-MAC instruction vs 8 v_cmp for the same pairs on VALU).
// CDNA5-specific path used instead: the Tensor Data Mover. One
// tensor_load_to_lds per workgroup DMAs the board's history row into LDS
// (TENSORcnt-tracked); the wait is split so the DMA overlaps all per-lane
// global loads of the candidate-hash construction. Inner loop is
// uniform-address ds_load_b128 broadcast + v_cmp.
// ---------------------------------------------------------------------------

typedef __attribute__((ext_vector_type(4))) unsigned int tdm_u32x4;
typedef __attribute__((ext_vector_type(8))) int          tdm_i32x8;
typedef __attribute__((ext_vector_type(4))) int          tdm_i32x4;

#define HIST_CAP 768  // >= M (722), 16B-aligned capacity for int4 LDS loads

// Kernel A: per-group XOR of per-stone Zobrist deltas.
// group_xor[g] = XOR_{s in [sgp[g],sgp[g+1])} ( Z[1+opp(board(g))][idx_s] ^ Z[0][idx_s] )
__global__ void go_group_xor_kernel(const int* __restrict__ ZposT,
                                    const int* __restrict__ current_player,
                                    const int* __restrict__ stone_global_index,
                                    const int* __restrict__ stone_global_pointer,
                                    const int* __restrict__ group_ptr_per_board,
                                    int* __restrict__ group_xor,
                                    int R, int B, int N2) {
  int g = blockIdx.x * blockDim.x + threadIdx.x;
  if (g >= R) return;
  // board(g): largest b with group_ptr_per_board[b] <= g  (matches jnp.repeat)
  int lo = 0, hi = B - 1;
  while (lo < hi) {
    int mid = (lo + hi + 1) >> 1;
    if (group_ptr_per_board[mid] <= g) lo = mid; else hi = mid - 1;
  }
  const int b   = lo;
  const int opp = 1 - current_player[b];
  const int* __restrict__ z_emp = ZposT;
  const int* __restrict__ z_opp = ZposT + (1 + opp) * N2;
  const int s0 = stone_global_pointer[g];
  const int s1 = stone_global_pointer[g + 1];
  int x = 0;
  for (int s = s0; s < s1; ++s) {
    const int idx = stone_global_index[s];
    x ^= z_opp[idx] ^ z_emp[idx];
  }
  group_xor[g] = x;
}

// Kernel B: one workgroup per board.
//   1) wave 0 launches the TDM DMA of hash_history[b,:] into LDS (no wait)
//   2) all lanes build their candidate hash from global memory (overlaps DMA)
//   3) wave 0 drains TENSORcnt, barrier publishes LDS
//   4) invalid slots -> INT32_MIN (exact reference semantics), barrier
//   5) each lane scans LDS history (b128 broadcast), emits legal & !repeat
__global__ __launch_bounds__(384) void go_superko_kernel(
    const int* __restrict__ legal_mask,          // nonzero == true
    const int* __restrict__ current_player,
    const int* __restrict__ current_hash,
    const int* __restrict__ hash_history,
    const int* __restrict__ move_count,
    const int* __restrict__ ZposT,
    const int* __restrict__ group_ptr_per_board,
    const int* __restrict__ captured_idx,        // (B, N2, 4) int32
    const int* __restrict__ group_xor,
    float* __restrict__ out,
    int M, int N2) {
  __shared__ __align__(16) int hist[HIST_CAP];
  const int b   = blockIdx.x;
  const int tid = threadIdx.x;

  // --- (1) TDM issue: 1-row tile of M 4-byte elements, global -> LDS.
  // Uniform branch so only wave 0's instruction stream reaches the TDM op
  // (TDM ignores EXEC; completion tracked by wave 0's TENSORcnt).
  // Descriptor per CDNA5 ISA sec. 8 (D# groups 0/1; groups 2..4 zero, <=2D).
  if (tid < 32) {
    const unsigned long long ga =
        (unsigned long long)(const void*)hash_history +
        (unsigned long long)b * (unsigned long long)M * 4ull;
    const unsigned lds_addr = (unsigned)(unsigned long long)(void*)&hist[0];

    tdm_u32x4 g0;
    g0[0] = 1u;                                  // count=1, user mode
    g0[1] = lds_addr;                            // lds_addr [63:32]
    g0[2] = (unsigned)ga;                        // global_addr [95:64]
    g0[3] = ((unsigned)(ga >> 32) & 0x1FFFFFFu)  // global_addr [120:96]
            | (2u << 30);                        // type=2 ("image")

    tdm_i32x8 g1;
    g1[0] = 0x00020000;                          // data_size=2 -> 4B; no mask/pad/iter
    g1[1] = (M & 0xFFFF) << 16;                  // tensor_dim0 lo16 (bits 63:48)
    g1[2] = ((M >> 16) & 0xFFFF) | (1 << 16);    // tensor_dim0 hi16 | tensor_dim1=1 lo16
    g1[3] = (M & 0xFFFF) << 16;                  // tensor_dim1 hi16=0 | tile_dim0=M
    g1[4] = 1;                                   // tile_dim1=1, tile_dim2=0
    g1[5] = M;                                   // tensor_dim0_stride lo32
    g1[6] = 0;                                   // stride0 hi16 | stride1 lo16
    g1[7] = 0;                                   // stride1 hi32

    tdm_i32x4 g2 = {0, 0, 0, 0};                 // tensor_dim2..: unused (2D)
    tdm_i32x4 g3 = {0, 0, 0, 0};                 // tensor_dim3..: unused
    tdm_i32x8 g4 = {0, 0, 0, 0, 0, 0, 0, 0};     // extra group (6-arg form)
    __builtin_amdgcn_tensor_load_to_lds(g0, g1, g2, g3, g4, 0);
    // NOTE: no wait here -- overlap the DMA with the candidate build below.
  }

  // --- (2) Candidate-hash construction: pure global-memory work, fully
  // independent of the in-flight DMA.
  int cand  = 0;
  int legal = 0;
  if (tid < N2) {
    const int n      = tid;
    const int player = current_player[b];
    // placement_delta = z_empty[n] ^ z_color[player][n]
    const int z_place = ZposT[n] ^ ZposT[(1 + player) * N2 + n];

    // capture_delta: 4 capture slots per point, one b128 load.
    const int4 cap  = ((const int4*)captured_idx)[(size_t)b * N2 + n];
    const int gbase = group_ptr_per_board[b];
    int cd = 0;
    if (cap.x >= 0) cd ^= group_xor[gbase + cap.x];
    if (cap.y >= 0) cd ^= group_xor[gbase + cap.y];
    if (cap.z >= 0) cd ^= group_xor[gbase + cap.z];
    if (cap.w >= 0) cd ^= group_xor[gbase + cap.w];

    cand  = current_hash[b] ^ z_place ^ cd;
    legal = legal_mask[(size_t)b * N2 + n];
  }
  int mp = move_count[b];
  if (mp > M) mp = M;

  // --- (3) Drain the DMA (issuing wave only), then publish LDS.
  if (tid < 32) __builtin_amdgcn_s_wait_tensorcnt(0);
  __syncthreads();

  // --- (4) Invalid history slots (t >= min(move_count, M)) compare as
  // INT32_MIN, exactly like the reference's jnp.where(valid, hist, INT32_MIN).
  for (int t = tid; t < M; t += (int)blockDim.x)
    if (t >= mp) hist[t] = (int)0x80000000;
  __syncthreads();

  // --- (5) Scan LDS history: uniform-address ds_load_b128 broadcast,
  // 4 cmps/16B. Exactly M entries compared (bit-exact vs reference).
  if (tid < N2) {
    int rep = 0;
    const int Mq = M & ~3;
    for (int j = 0; j < Mq; j += 4) {
      const int4 q = *(const int4*)&hist[j];
      rep |= (cand == q.x);
      rep |= (cand == q.y);
      rep |= (cand == q.z);
      rep |= (cand == q.w);
    }
    for (int j = Mq; j < M; ++j) rep |= (cand == hist[j]);

    out[(size_t)b * N2 + tid] = (legal != 0 && rep == 0) ? 1.0f : 0.0f;
  }
}

extern "C" void kernel_launch(void* const* d_in, const int* in_sizes, int n_in,
                              void* d_out, int out_size, void* d_ws, size_t ws_size,
                              hipStream_t stream) {
  (void)n_in; (void)out_size; (void)ws_size;
  // setup_inputs() order:
  const int* legal  = (const int*)d_in[0];  // legal_mask (bool -> int per harness)
  const int* cplay  = (const int*)d_in[1];  // current_player
  const int* chash  = (const int*)d_in[2];  // current_hash
  const int* hhist  = (const int*)d_in[3];  // hash_history (B, M)
  const int* mcount = (const int*)d_in[4];  // move_count
  const int* zpos   = (const int*)d_in[5];  // ZposT (3, N2)
  const int* sgi    = (const int*)d_in[6];  // stone_global_index (K)
  const int* sgp    = (const int*)d_in[7];  // stone_global_pointer (R+1)
  const int* gpp    = (const int*)d_in[8];  // group_global_pointer_per_board (B+1)
  const int* cgli   = (const int*)d_in[9];  // captured_group_local_index (B, N2, 4)

  const int B  = in_sizes[8] - 1;
  const int R  = in_sizes[7] - 1;
  const int M  = in_sizes[3] / B;   // 722
  const int N2 = in_sizes[5] / 3;   // 361

  int* gx = (int*)d_ws;             // R * 4 bytes of scratch (32 KB)

  go_group_xor_kernel<<<(R + 255) / 256, 256, 0, stream>>>(
      zpos, cplay, sgi, sgp, gpp, gx, R, B, N2);

  go_superko_kernel<<<B, 384, 0, stream>>>(
      legal, cplay, chash, hhist, mcount, zpos, gpp, cgli, gx,
      (float*)d_out, M, N2);
}